// GAT_61564061221033
// MI455X (gfx1250) — compile-verified
//
#include <hip/hip_runtime.h>
#include <hip/hip_bf16.h>

// ---------------------------------------------------------------------------
// 2-layer GAT forward (fp32), gfx1250. GEMMs via V_WMMA_F32_16X16X4_F32.
// ---------------------------------------------------------------------------

typedef float v2f __attribute__((ext_vector_type(2)));
typedef float v8f __attribute__((ext_vector_type(8)));

#define F_DIM 128      // F_IN == HEADS*HID == HEADS*OUT == 128
#define HEADS 2
#define HALF_F 64
#define NEG_SLOPE 0.2f
#define BN_EPS 1e-5f

__device__ __forceinline__ float leaky(float x) {
    return x >= 0.f ? x : NEG_SLOPE * x;
}
// monotonic uint mapping for float atomicMax (handles negatives)
__device__ __forceinline__ unsigned f2mono(float f) {
    unsigned u = __float_as_uint(f);
    return (u & 0x80000000u) ? ~u : (u | 0x80000000u);
}
__device__ __forceinline__ float mono2f(unsigned u) {
    return (u & 0x80000000u) ? __uint_as_float(u & 0x7FFFFFFFu)
                             : __uint_as_float(~u);
}

// ---------------------------------------------------------------------------
// GEMM: C(nrows x 128) = A(nrows x 128) @ B(128 x 128), fp32 WMMA.
// blockDim = 256 (8 waves). Each block: one 16-row M tile; wave w -> N tile w.
// nrows must be a multiple of 16 (100000 = 16*6250).
// ---------------------------------------------------------------------------
__global__ void gat_gemm128_wmma(const float* __restrict__ A,
                                 const float* __restrict__ B,
                                 float* __restrict__ C) {
    const int lane = threadIdx.x & 31;
    const int wv   = threadIdx.x >> 5;          // 0..7 -> N tile
    const int half = lane >> 4;                 // 0 or 1
    const int l    = lane & 15;
    const int row0 = blockIdx.x * 16;
    const int col0 = wv * 16;

    v8f acc = {};
#pragma unroll
    for (int k0 = 0; k0 < F_DIM; k0 += 4) {
        // A frag: this lane holds A[row0+l][k0+2*half .. +1]
        v2f a = *(const v2f*)(A + (size_t)(row0 + l) * F_DIM + k0 + 2 * half);
        // B frag: VGPR0 = B[k0+2*half][col0+l], VGPR1 = B[k0+2*half+1][col0+l]
        v2f b;
        b.x = B[(size_t)(k0 + 2 * half + 0) * F_DIM + col0 + l];
        b.y = B[(size_t)(k0 + 2 * half + 1) * F_DIM + col0 + l];
        acc = __builtin_amdgcn_wmma_f32_16x16x4_f32(
            false, a, false, b, (short)0, acc, false, false);
    }
    // C frag: VGPR r -> row (r + 8*half), col l
#pragma unroll
    for (int r = 0; r < 8; ++r) {
        C[(size_t)(row0 + 8 * half + r) * F_DIM + col0 + l] = acc[r];
    }
}

// ---------------------------------------------------------------------------
// Per-node attention coefficients: a_src[n,h] = dot(h[n, h*64:...], att_src[h])
// One wave per node: lanes 0-15 -> head 0, lanes 16-31 -> head 1 (4 f32/lane).
// ---------------------------------------------------------------------------
__global__ void gat_attn_coef(const float* __restrict__ h,
                              const float* __restrict__ att_src,
                              const float* __restrict__ att_dst,
                              float* __restrict__ asrc,
                              float* __restrict__ adst, int n) {
    const int node = blockIdx.x * (blockDim.x >> 5) + (threadIdx.x >> 5);
    const int lane = threadIdx.x & 31;
    if (node >= n) return;
    const int head = lane >> 4;
    const int l    = lane & 15;
    const float* hp = h + (size_t)node * F_DIM + head * HALF_F + l * 4;
    const float* sp = att_src + head * HALF_F + l * 4;
    const float* dp = att_dst + head * HALF_F + l * 4;
    float s = 0.f, d = 0.f;
#pragma unroll
    for (int i = 0; i < 4; ++i) {
        float v = hp[i];
        s += v * sp[i];
        d += v * dp[i];
    }
#pragma unroll
    for (int off = 8; off >= 1; off >>= 1) {
        s += __shfl_xor(s, off, 32);
        d += __shfl_xor(d, off, 32);
    }
    if (l == 0) {
        asrc[(size_t)node * HEADS + head] = s;
        adst[(size_t)node * HEADS + head] = d;
    }
}

__global__ void gat_zero_f32(float* __restrict__ p, int n) {
    int gid = blockIdx.x * blockDim.x + threadIdx.x;
    if (gid < n) p[gid] = 0.f;
}

// init running max with the self-loop logit (every dst has a self loop),
// zero softmax denominator.
__global__ void gat_attn_init(const float* __restrict__ asrc,
                              const float* __restrict__ adst,
                              unsigned* __restrict__ mkey,
                              float* __restrict__ denom, int nh) {
    int gid = blockIdx.x * blockDim.x + threadIdx.x;
    if (gid >= nh) return;
    float es  = leaky(asrc[gid] + adst[gid]);
    mkey[gid] = f2mono(es);
    denom[gid] = 0.f;
}

// segment max over dst (real edges)
__global__ void gat_edge_max(const int* __restrict__ ei, int E,
                             const float* __restrict__ asrc,
                             const float* __restrict__ adst,
                             unsigned* __restrict__ mkey) {
    int gid = blockIdx.x * blockDim.x + threadIdx.x;
    if (gid >= E * HEADS) return;
    int e = gid >> 1, h = gid & 1;
    int s = ei[e], d = ei[E + e];
    float v = leaky(asrc[(size_t)s * HEADS + h] + adst[(size_t)d * HEADS + h]);
    atomicMax(&mkey[(size_t)d * HEADS + h], f2mono(v));
}

// segment sum of exp(e - m) over dst (real edges)
__global__ void gat_edge_sum(const int* __restrict__ ei, int E,
                             const float* __restrict__ asrc,
                             const float* __restrict__ adst,
                             const unsigned* __restrict__ mkey,
                             float* __restrict__ denom) {
    int gid = blockIdx.x * blockDim.x + threadIdx.x;
    if (gid >= E * HEADS) return;
    int e = gid >> 1, h = gid & 1;
    int s = ei[e], d = ei[E + e];
    float v = leaky(asrc[(size_t)s * HEADS + h] + adst[(size_t)d * HEADS + h]);
    float m = mono2f(mkey[(size_t)d * HEADS + h]);
    atomicAdd(&denom[(size_t)d * HEADS + h], expf(v - m));
}

// add self-loop term to denom (separate launch -> no atomics needed)
__global__ void gat_self_denom(const float* __restrict__ asrc,
                               const float* __restrict__ adst,
                               const unsigned* __restrict__ mkey,
                               float* __restrict__ denom, int nh) {
    int gid = blockIdx.x * blockDim.x + threadIdx.x;
    if (gid >= nh) return;
    float es = leaky(asrc[gid] + adst[gid]);
    denom[gid] += expf(es - mono2f(mkey[gid]));
}

// accum[d, f] += h[s, f] * alpha(e, head(f)); one thread per (edge, feature)
__global__ void gat_scatter(const int* __restrict__ ei, int E,
                            const float* __restrict__ h,
                            const float* __restrict__ asrc,
                            const float* __restrict__ adst,
                            const unsigned* __restrict__ mkey,
                            const float* __restrict__ denom,
                            float* __restrict__ accum) {
    int gid = blockIdx.x * blockDim.x + threadIdx.x;
    if (gid >= E * F_DIM) return;
    int e = gid >> 7;            // /128
    int f = gid & (F_DIM - 1);
    int hd = f >> 6;             // head
    int s = ei[e], d = ei[E + e];
    size_t ih = (size_t)d * HEADS + hd;
    float v = leaky(asrc[(size_t)s * HEADS + hd] + adst[ih]);
    float alpha = expf(v - mono2f(mkey[ih])) / denom[ih];
    atomicAdd(&accum[(size_t)d * F_DIM + f],
              h[(size_t)s * F_DIM + f] * alpha);
}

// layer-1 finalize: add self-loop contribution + bias, concat heads
__global__ void gat_finalize1(const float* __restrict__ accum,
                              const float* __restrict__ h,
                              const float* __restrict__ asrc,
                              const float* __restrict__ adst,
                              const unsigned* __restrict__ mkey,
                              const float* __restrict__ denom,
                              const float* __restrict__ bias,
                              float* __restrict__ out, int n) {
    int gid = blockIdx.x * blockDim.x + threadIdx.x;
    if (gid >= n * F_DIM) return;
    int node = gid >> 7;
    int f    = gid & (F_DIM - 1);
    int hd   = f >> 6;
    size_t ih = (size_t)node * HEADS + hd;
    float es = leaky(asrc[ih] + adst[ih]);
    float aself = expf(es - mono2f(mkey[ih])) / denom[ih];
    out[gid] = accum[gid] + h[gid] * aself + bias[f];
}

// BN stats: blockDim=128 (one thread per channel); each block reduces a chunk
// of rows and atomically accumulates per-channel sum / sum-of-squares.
__global__ void gat_bn_stats(const float* __restrict__ h, int nrows,
                             float* __restrict__ sum,
                             float* __restrict__ sumsq) {
    int c = threadIdx.x;
    int r0 = blockIdx.x * 256;
    int r1 = min(r0 + 256, nrows);
    float s = 0.f, q = 0.f;
    for (int r = r0; r < r1; ++r) {
        float v = h[(size_t)r * F_DIM + c];
        s += v;
        q += v * v;
    }
    atomicAdd(&sum[c], s);
    atomicAdd(&sumsq[c], q);
}

// in-place BN (batch stats, biased var) + ELU
__global__ void gat_bn_elu(float* __restrict__ h, int n,
                           const float* __restrict__ sum,
                           const float* __restrict__ sumsq,
                           const float* __restrict__ gamma,
                           const float* __restrict__ beta) {
    int gid = blockIdx.x * blockDim.x + threadIdx.x;
    if (gid >= n * F_DIM) return;
    int c = gid & (F_DIM - 1);
    float inv_n = 1.f / (float)n;
    float mu  = sum[c] * inv_n;
    float var = sumsq[c] * inv_n - mu * mu;
    float y = (h[gid] - mu) * rsqrtf(var + BN_EPS) * gamma[c] + beta[c];
    h[gid] = y > 0.f ? y : expm1f(y);
}

// layer-2 finalize: self-loop + head mean + bias -> (N, 64) output
__global__ void gat_finalize2(const float* __restrict__ accum,
                              const float* __restrict__ h,
                              const float* __restrict__ asrc,
                              const float* __restrict__ adst,
                              const unsigned* __restrict__ mkey,
                              const float* __restrict__ denom,
                              const float* __restrict__ bias,
                              float* __restrict__ out, int n) {
    int gid = blockIdx.x * blockDim.x + threadIdx.x;
    if (gid >= n * HALF_F) return;
    int node = gid >> 6;
    int f    = gid & (HALF_F - 1);
    float acc = 0.f;
#pragma unroll
    for (int hd = 0; hd < HEADS; ++hd) {
        size_t ih = (size_t)node * HEADS + hd;
        float es = leaky(asrc[ih] + adst[ih]);
        float aself = expf(es - mono2f(mkey[ih])) / denom[ih];
        size_t off = (size_t)node * F_DIM + hd * HALF_F + f;
        acc += accum[off] + h[off] * aself;
    }
    out[gid] = acc * (1.f / HEADS) + bias[f];
}

// ---------------------------------------------------------------------------
extern "C" void kernel_launch(void* const* d_in, const int* in_sizes, int n_in,
                              void* d_out, int out_size, void* d_ws,
                              size_t ws_size, hipStream_t stream) {
    const float* x        = (const float*)d_in[0];
    const int*   ei       = (const int*)d_in[1];   // [2, E]
    const float* W1       = (const float*)d_in[2];
    const float* att_src1 = (const float*)d_in[3];
    const float* att_dst1 = (const float*)d_in[4];
    const float* bias1    = (const float*)d_in[5];
    const float* gamma    = (const float*)d_in[6];
    const float* beta     = (const float*)d_in[7];
    const float* W2       = (const float*)d_in[8];
    const float* att_src2 = (const float*)d_in[9];
    const float* att_dst2 = (const float*)d_in[10];
    const float* bias2    = (const float*)d_in[11];
    float* out = (float*)d_out;

    const int N = in_sizes[0] / F_DIM;   // 100000
    const int E = in_sizes[1] / 2;       // 800000
    const int NH = N * HEADS;

    // workspace layout
    char* base = (char*)d_ws;
    const size_t nb = (size_t)N * F_DIM * sizeof(float);
    float*    h     = (float*)(base);                 // N x 128
    float*    accum = (float*)(base + nb);            // N x 128
    float*    hbn   = (float*)(base + 2 * nb);        // N x 128
    float*    asrc  = (float*)(base + 3 * nb);
    float*    adst  = asrc + NH;
    unsigned* mkey  = (unsigned*)(adst + NH);
    float*    denom = (float*)(mkey + NH);
    float*    bsum  = denom + NH;                     // 128
    float*    bsq   = bsum + F_DIM;                   // 128 (contiguous)

    const int TB = 256;
    const int gNode   = (N + 7) / 8;                  // 8 waves/block in coef
    const int gNH     = (NH + TB - 1) / TB;
    const int gEH     = (E * HEADS + TB - 1) / TB;
    const int gEF     = (E * F_DIM + TB - 1) / TB;
    const int gNF     = (N * F_DIM + TB - 1) / TB;
    const int gOut    = (N * HALF_F + TB - 1) / TB;
    const int gBN     = (N + 255) / 256;

    // ---------------- layer 1 ----------------
    gat_gemm128_wmma<<<N / 16, TB, 0, stream>>>(x, W1, h);
    gat_attn_coef<<<gNode, TB, 0, stream>>>(h, att_src1, att_dst1, asrc, adst, N);
    gat_attn_init<<<gNH, TB, 0, stream>>>(asrc, adst, mkey, denom, NH);
    gat_zero_f32<<<gNF, TB, 0, stream>>>(accum, N * F_DIM);
    gat_edge_max<<<gEH, TB, 0, stream>>>(ei, E, asrc, adst, mkey);
    gat_edge_sum<<<gEH, TB, 0, stream>>>(ei, E, asrc, adst, mkey, denom);
    gat_self_denom<<<gNH, TB, 0, stream>>>(asrc, adst, mkey, denom, NH);
    gat_scatter<<<gEF, TB, 0, stream>>>(ei, E, h, asrc, adst, mkey, denom, accum);
    gat_finalize1<<<gNF, TB, 0, stream>>>(accum, h, asrc, adst, mkey, denom,
                                          bias1, hbn, N);

    // BatchNorm (batch stats) + ELU, in place on hbn
    gat_zero_f32<<<1, 256, 0, stream>>>(bsum, 2 * F_DIM);
    gat_bn_stats<<<gBN, F_DIM, 0, stream>>>(hbn, N, bsum, bsq);
    gat_bn_elu<<<gNF, TB, 0, stream>>>(hbn, N, bsum, bsq, gamma, beta);

    // ---------------- layer 2 ----------------
    gat_gemm128_wmma<<<N / 16, TB, 0, stream>>>(hbn, W2, h);
    gat_attn_coef<<<gNode, TB, 0, stream>>>(h, att_src2, att_dst2, asrc, adst, N);
    gat_attn_init<<<gNH, TB, 0, stream>>>(asrc, adst, mkey, denom, NH);
    gat_zero_f32<<<gNF, TB, 0, stream>>>(accum, N * F_DIM);
    gat_edge_max<<<gEH, TB, 0, stream>>>(ei, E, asrc, adst, mkey);
    gat_edge_sum<<<gEH, TB, 0, stream>>>(ei, E, asrc, adst, mkey, denom);
    gat_self_denom<<<gNH, TB, 0, stream>>>(asrc, adst, mkey, denom, NH);
    gat_scatter<<<gEF, TB, 0, stream>>>(ei, E, h, asrc, adst, mkey, denom, accum);
    gat_finalize2<<<gOut, TB, 0, stream>>>(accum, h, asrc, adst, mkey, denom,
                                           bias2, out, N);
}